// ElevationSConvResidualEncoder_17789754540232
// MI455X (gfx1250) — compile-verified
//
#include <hip/hip_runtime.h>
#include <hip/hip_bf16.h>
#include <stdint.h>

// Problem geometry (fixed by the reference)
#define B_      8
#define C1_     64
#define C2_     64
#define T_      1020
#define POOL_   12
#define TP_     85          // T_/POOL_
#define HID_    512
#define DIM_    256
#define OC_     4
#define COLS_   (B_*C1_)            // 512 independent (b,x) columns
#define FRAME_  (B_*C1_*C2_)        // 32768 elements per pooled timestep

typedef float v2f __attribute__((ext_vector_type(2)));
typedef float v8f __attribute__((ext_vector_type(8)));

// Fast activations: keep the serial ConvLSTM critical path on single
// hardware trans ops (v_tanh_f32 / v_exp_f32 / v_rcp_f32), never the
// 12-op IEEE divide expansion.
__device__ __forceinline__ float fast_tanh(float x) {
#if __has_builtin(__builtin_amdgcn_tanhf)
    return __builtin_amdgcn_tanhf(x);           // v_tanh_f32
#else
    float cx = fminf(fmaxf(x, -15.0f), 15.0f);
    float e  = __expf(2.0f * cx);               // v_exp_f32
    return (e - 1.0f) * __builtin_amdgcn_rcpf(e + 1.0f);   // v_rcp_f32
#endif
}
__device__ __forceinline__ float fast_sigmoid(float x) {
#if __has_builtin(__builtin_amdgcn_tanhf)
    return fmaf(0.5f, __builtin_amdgcn_tanhf(0.5f * x), 0.5f);
#else
    return __builtin_amdgcn_rcpf(1.0f + __expf(-x));
#endif
}

// ---------------------------------------------------------------------------
// Phase 1: temporal avg-pool (kernel=stride=12) + transpose to [t][b][x][y].
// Streams the 133MB input once (the compulsory HBM traffic, ~5.7us at
// 23.3TB/s); 16B-aligned float4 loads; coalesced stores; L2 absorbs the
// cross-line sharing. Block 0 zeroes the 32-float spike accumulator.
// ---------------------------------------------------------------------------
__global__ void __launch_bounds__(256)
k_pool(const float* __restrict__ spikes, float* __restrict__ reduced,
       float* __restrict__ pooled_acc) {
    if (blockIdx.x == 0 && threadIdx.x < B_ * OC_) pooled_acc[threadIdx.x] = 0.0f;
    int o = blockIdx.x * 256 + threadIdx.x;
    if (o >= TP_ * FRAME_) return;
    int tp = o >> 15;            // o / 32768
    int r  = o & 32767;          // b*4096 + x*64 + y == input row id
    const float4* p = (const float4*)(spikes + (size_t)r * T_ + tp * POOL_);
    float4 a = p[0], b = p[1], c = p[2];
    float s = (a.x + a.y + a.z + a.w) + (b.x + b.y + b.z + b.w)
            + (c.x + c.y + c.z + c.w);
    reduced[o] = s * (1.0f / (float)POOL_);
}

// ---------------------------------------------------------------------------
// Phase 2: ConvLSTM recurrence. Conv kernel is (3,1): coupling only along y,
// so the 512 (b,x) columns are independent. One 64-thread block (2 waves)
// per column, one thread per y. Membrane state exchanged via LDS (halo=0
// implements the conv zero padding). The frame column for step t+1 is
// prefetched into an LDS double buffer with GLOBAL_LOAD_ASYNC_TO_LDS_B32
// (ASYNCcnt-tracked) while step t computes, hiding global latency on the
// 85-step serial critical path.
// ---------------------------------------------------------------------------
__global__ void __launch_bounds__(64)
k_lstm(const float* __restrict__ reduced,
       const float* __restrict__ convW,   // [16][5][3]
       const float* __restrict__ convB,   // [16]
       float* __restrict__ pooled_acc) {  // [B][OC]
    __shared__ __align__(16) float wsh[16 * 16];   // gate rows padded to 16
    __shared__ float memsh[OC_][C2_ + 2];          // halo at [0] and [65]
    __shared__ float frsh[2][C2_ + 2];             // double-buffered frame
    __shared__ float redsh[OC_];

    const int ty  = threadIdx.x;       // y
    const int col = blockIdx.x;        // b*64 + x

    if (ty < 16) {
        #pragma unroll
        for (int k = 0; k < 15; ++k) wsh[ty * 16 + k] = convW[ty * 15 + k];
        wsh[ty * 16 + 15] = 0.0f;
    }
    #pragma unroll
    for (int c = 0; c < OC_; ++c) {
        memsh[c][ty + 1] = 0.0f;
        if (ty < 2) memsh[c][ty * (C2_ + 1)] = 0.0f;   // halos
    }
    if (ty < 2) { frsh[ty][0] = 0.0f; frsh[ty][C2_ + 1] = 0.0f; }
    if (ty < OC_) redsh[ty] = 0.0f;

    // Gate biases: uniform, read once into registers (saves 16 DS broadcast
    // reads per step on the critical path).
    float bias_r[16];
    #pragma unroll
    for (int o = 0; o < 16; ++o) bias_r[o] = convB[o];

    // LDS byte offsets of this thread's frame slots (generic LDS pointer:
    // low 32 bits are the wave-relative LDS offset).
    unsigned lds0 = (unsigned)(size_t)&frsh[0][1 + ty];
    unsigned lds1 = (unsigned)(size_t)&frsh[1][1 + ty];
    const float* gbase = reduced;
    int base_off = (col * C2_ + ty) * 4;   // byte offset within one frame

    // Prefetch frame t=0 into buffer 0 (async, tracked by ASYNCcnt).
    asm volatile("global_load_async_to_lds_b32 %0, %1, %2 offset:0"
                 :: "v"(lds0), "v"(base_off), "s"(gbase) : "memory");

    float syn[OC_]  = {0.f, 0.f, 0.f, 0.f};
    float mem[OC_]  = {0.f, 0.f, 0.f, 0.f};
    float sacc[OC_] = {0.f, 0.f, 0.f, 0.f};

    __syncthreads();

    for (int t = 0; t < TP_; ++t) {
        if (t + 1 < TP_) {   // kick prefetch of t+1, then wait for t's data
            unsigned ldsn = ((t + 1) & 1) ? lds1 : lds0;
            int offn = (t + 1) * (FRAME_ * 4) + base_off;
            asm volatile("global_load_async_to_lds_b32 %0, %1, %2 offset:0"
                         :: "v"(ldsn), "v"(offn), "s"(gbase) : "memory");
            asm volatile("s_wait_asynccnt 1" ::: "memory");
        } else {
            asm volatile("s_wait_asynccnt 0" ::: "memory");
        }
        __syncthreads();   // frame(t) + memsh(t-1) visible to all threads

        const float* frp = (t & 1) ? frsh[1] : frsh[0];
        float fr0 = frp[ty], fr1 = frp[ty + 1], fr2 = frp[ty + 2];
        float m0[OC_], m1[OC_], m2[OC_];
        #pragma unroll
        for (int c = 0; c < OC_; ++c) {
            m0[c] = memsh[c][ty];
            m1[c] = memsh[c][ty + 1];
            m2[c] = memsh[c][ty + 2];
        }

        auto gate = [&](int o) -> float {
            const float4* w = (const float4*)(wsh + o * 16);
            float4 w0 = w[0], w1 = w[1], w2 = w[2], w3 = w[3];
            float acc = bias_r[o];
            acc = fmaf(w0.x, fr0, acc); acc = fmaf(w0.y, fr1, acc); acc = fmaf(w0.z, fr2, acc);
            acc = fmaf(w0.w, m0[0], acc); acc = fmaf(w1.x, m1[0], acc); acc = fmaf(w1.y, m2[0], acc);
            acc = fmaf(w1.z, m0[1], acc); acc = fmaf(w1.w, m1[1], acc); acc = fmaf(w2.x, m2[1], acc);
            acc = fmaf(w2.y, m0[2], acc); acc = fmaf(w2.z, m1[2], acc); acc = fmaf(w2.w, m2[2], acc);
            acc = fmaf(w3.x, m0[3], acc); acc = fmaf(w3.y, m1[3], acc); acc = fmaf(w3.z, m2[3], acc);
            return acc;
        };

        #pragma unroll
        for (int c = 0; c < OC_; ++c) {
            float gi = gate(c), gf = gate(4 + c), gg = gate(8 + c), go = gate(12 + c);
            syn[c] = fast_sigmoid(gf) * syn[c] + fast_sigmoid(gi) * fast_tanh(gg);
            mem[c] = fast_sigmoid(go) * fast_tanh(syn[c]);
            sacc[c] += (mem[c] > 1.0f) ? 1.0f : 0.0f;   // spike, threshold=1, no reset
        }

        __syncthreads();   // all reads of memsh(t-1) done
        #pragma unroll
        for (int c = 0; c < OC_; ++c) memsh[c][ty + 1] = mem[c];
    }

    __syncthreads();
    #pragma unroll
    for (int c = 0; c < OC_; ++c) atomicAdd(&redsh[c], sacc[c]);
    __syncthreads();
    if (ty < OC_) atomicAdd(&pooled_acc[(col >> 6) * OC_ + ty], redsh[ty]);
}

// ---------------------------------------------------------------------------
// Phase 3: three [8,256]x[256,512] GEMMs via V_WMMA_F32_16X16X4_F32 (M padded
// 8->16 with cndmask-zeroed rows, no EXEC divergence at the WMMA). One wave
// per (matrix, 16-col tile): 3*32 = 96 waves, 64 WMMAs each over K=256.
// Elevation epilogue fuses the rank-4 spike residual + gain + double ReLU.
// ---------------------------------------------------------------------------
__global__ void __launch_bounds__(32)
k_gemm(const float* __restrict__ Xd, const float* __restrict__ Xa,
       const float* __restrict__ Xe,
       const float* __restrict__ Wd, const float* __restrict__ bd,
       const float* __restrict__ Wa, const float* __restrict__ ba,
       const float* __restrict__ We, const float* __restrict__ be,
       const float* __restrict__ projW, const float* __restrict__ projb,
       const float* __restrict__ rg, const float* __restrict__ pooled_acc,
       float* __restrict__ out) {
    int mat  = blockIdx.x >> 5;
    int tile = blockIdx.x & 31;
    const float* X    = (mat == 0) ? Xd : (mat == 1) ? Xa : Xe;
    const float* W    = (mat == 0) ? Wd : (mat == 1) ? Wa : We;
    const float* bias = (mat == 0) ? bd : (mat == 1) ? ba : be;

    int lane = threadIdx.x;
    int hi   = lane >> 4;           // lane group (0: K j, 1: K j+2 / M j+8)
    int lo   = lane & 15;           // M for A, N for B/C
    int row  = lane & 7;            // clamped batch row (pad rows read row&7)
    bool rowok = lo < 8;
    int  colc  = tile * 16 + lo;

    v8f acc = {};
    for (int kk = 0; kk < DIM_; kk += 4) {
        int k0 = kk + 2 * hi;
        float a0 = X[row * DIM_ + k0];
        float a1 = X[row * DIM_ + k0 + 1];
        v2f a, b;
        a.x = rowok ? a0 : 0.0f;            // pad M=8..15 with zeros
        a.y = rowok ? a1 : 0.0f;
        b.x = W[colc * DIM_ + k0];          // B[k][n] = W[n][k]
        b.y = W[colc * DIM_ + k0 + 1];
        acc = __builtin_amdgcn_wmma_f32_16x16x4_f32(
                  false, a, false, b, (short)0, acc, false, false);
    }

    float scale = 0.0f;
    if (mat == 2) scale = 0.4f * fast_sigmoid(rg[0]);
    const float inv_cnt = 1.0f / (float)(TP_ * C2_ * C1_);

    #pragma unroll
    for (int j = 0; j < 8; ++j) {
        int mrow = j + 8 * hi;              // C/D layout: VGPR j -> M=j (+8 hi)
        if (mrow < 8) {
            float v = fmaxf(acc[j] + bias[colc], 0.0f);   // relu(x@W.T + b)
            if (mat == 2) {
                float r = projb[colc];
                #pragma unroll
                for (int c = 0; c < OC_; ++c)
                    r = fmaf(pooled_acc[mrow * OC_ + c] * inv_cnt,
                             projW[colc * OC_ + c], r);
                v = fmaxf(v + scale * r, 0.0f);           // relu(base + s*res)
            }
            out[mat * (B_ * HID_) + mrow * HID_ + colc] = v;
        }
    }
}

// ---------------------------------------------------------------------------
extern "C" void kernel_launch(void* const* d_in, const int* in_sizes, int n_in,
                              void* d_out, int out_size, void* d_ws, size_t ws_size,
                              hipStream_t stream) {
    const float* distance  = (const float*)d_in[0];
    const float* azimuth   = (const float*)d_in[1];
    const float* elevation = (const float*)d_in[2];
    const float* spikes    = (const float*)d_in[3];
    // d_in[4] spike_count: unused by the reference
    const float* Wd = (const float*)d_in[5];
    const float* bd = (const float*)d_in[6];
    const float* Wa = (const float*)d_in[7];
    const float* ba = (const float*)d_in[8];
    const float* We = (const float*)d_in[9];
    const float* be = (const float*)d_in[10];
    const float* convW = (const float*)d_in[11];
    const float* convB = (const float*)d_in[12];
    const float* projW = (const float*)d_in[13];
    const float* projb = (const float*)d_in[14];
    const float* rg    = (const float*)d_in[15];
    float* out = (float*)d_out;

    // Workspace: reduced [85][8][64][64] (11.14MB) then pooled_acc [8][4].
    float* reduced    = (float*)d_ws;
    float* pooled_acc = reduced + (size_t)TP_ * FRAME_;

    int nOut = TP_ * FRAME_;                     // 2,785,280 (exact multiple of 256)
    k_pool<<<nOut / 256, 256, 0, stream>>>(spikes, reduced, pooled_acc);
    k_lstm<<<COLS_, 64, 0, stream>>>(reduced, convW, convB, pooled_acc);
    k_gemm<<<96, 32, 0, stream>>>(distance, azimuth, elevation,
                                  Wd, bd, Wa, ba, We, be,
                                  projW, projb, rg, pooled_acc, out);
}